// DogNetwork_81956565942314
// MI455X (gfx1250) — compile-verified
//
#include <hip/hip_runtime.h>

typedef float v2f __attribute__((ext_vector_type(2)));
typedef float v8f __attribute__((ext_vector_type(8)));

#define WMMA_F32X4(a, b, c) \
  __builtin_amdgcn_wmma_f32_16x16x4_f32(false, (a), false, (b), (short)0, (c), false, false)

struct TP { const float* t[34]; };   // 34 leaves per transformer param dict

struct KArgs {
  const float* dog; const float* man; const float* hp; const float* mp;
  TP p1, p2, p3;
  const float* l1w; const float* l1b; const float* l2w; const float* l2b;
  float* out;
  int nrows;
};

// ---------------- per-lane (row) helpers: D <= 8 ----------------

// attn with seq_len==1: out = (x @ Wv^T + bv) @ Wo^T + bo
template<int D>
__device__ inline void attn_row(const float* x, const float* __restrict__ Wv,
                                const float* __restrict__ bv,
                                const float* __restrict__ Wo,
                                const float* __restrict__ bo, float* out) {
  float tmp[D];
  #pragma unroll
  for (int c = 0; c < D; ++c) {
    float s = bv[c];
    #pragma unroll
    for (int k = 0; k < D; ++k) s += x[k] * Wv[c * D + k];
    tmp[c] = s;
  }
  #pragma unroll
  for (int c = 0; c < D; ++c) {
    float s = bo[c];
    #pragma unroll
    for (int k = 0; k < D; ++k) s += tmp[k] * Wo[c * D + k];
    out[c] = s;
  }
}

template<int D>
__device__ inline void ln_row(float* x, const float* __restrict__ w,
                              const float* __restrict__ b) {
  float m = 0.0f;
  #pragma unroll
  for (int c = 0; c < D; ++c) m += x[c];
  m *= (1.0f / D);
  float v = 0.0f;
  #pragma unroll
  for (int c = 0; c < D; ++c) { float t = x[c] - m; v += t * t; }
  v *= (1.0f / D);
  const float s = rsqrtf(v + 1e-5f);
  #pragma unroll
  for (int c = 0; c < D; ++c) x[c] = (x[c] - m) * s * w[c] + b[c];
}

// ---------------- WMMA feed-forward: out = relu(h @ W1^T + b1) @ W2^T + b2 ----------------
// One wave handles a 16-row tile. ldsX: 16x8 f32 staging, ldsZ: 16x260 f32 intermediate.
// All OOB-avoidance is done by clamped indices (loads stay in-bounds & finite); the
// matching A elements are exact zeros, so the products vanish without EXEC masking.

template<int D>
__device__ void ff16(const float* h, const float* __restrict__ W1,
                     const float* __restrict__ b1, const float* __restrict__ W2,
                     const float* __restrict__ b2, float* out,
                     float* ldsX, float* ldsZ, int lane) {
  const int row   = lane & 15;        // M index (A) or N index (B/C)
  const int half  = lane >> 4;        // 0: lanes 0-15, 1: lanes 16-31
  const int khalf = half * 2;         // K sub-offset per ISA A/B layout
  constexpr int KP = (D + 3) & ~3;    // K padded to multiple of 4
  constexpr int NK = KP / 4;          // number of 16x16x4 k-steps in GEMM1
  constexpr int ZS = 260;             // Z row stride (banks split between halves)

  // stage activations, zero-padded to 8 columns
  if (lane < 16) {
    #pragma unroll
    for (int c = 0; c < 8; ++c) ldsX[lane * 8 + c] = (c < D) ? h[c] : 0.0f;
  }
  __syncthreads();

  // A operands are N-tile invariant: hoist out of the nt loop
  v2f aop[NK];
  #pragma unroll
  for (int i = 0; i < NK; ++i)
    aop[i] = *reinterpret_cast<const v2f*>(&ldsX[row * 8 + i * 4 + khalf]);

  // GEMM1: Z[16,256] = relu(X[16,D] @ W1^T + b1), W1 is [256, D] row-major
  for (int nt = 0; nt < 16; ++nt) {
    const int nb = nt * 16;
    const float* __restrict__ w1r = W1 + (nb + row) * D;   // B column nb+row
    v8f acc = {};
    #pragma unroll
    for (int i = 0; i < NK; ++i) {
      const int k0 = i * 4 + khalf, k1 = k0 + 1;
      const int k0c = (k0 < D) ? k0 : (D - 1);   // clamp: A is 0 there anyway
      const int k1c = (k1 < D) ? k1 : (D - 1);
      v2f b;
      b.x = w1r[k0c];
      b.y = w1r[k1c];
      acc = WMMA_F32X4(aop[i], b, acc);
    }
    const float bias = b1[nb + row];
    #pragma unroll
    for (int r = 0; r < 8; ++r) {
      const int m = r + half * 8;
      float v = acc[r] + bias;
      ldsZ[m * ZS + nb + row] = v > 0.0f ? v : 0.0f;
    }
  }
  __syncthreads();

  // GEMM2: O[16,D(pad 16)] = Z[16,256] @ W2^T, W2 is [D, 256] row-major.
  // Columns n >= D of the accumulator are garbage and never read.
  const int nc = (row < D) ? row : (D - 1);
  const float* __restrict__ w2p = W2 + nc * 256 + khalf;
  const float* zp = ldsZ + row * ZS + khalf;
  v8f acc = {};
  #pragma unroll 4
  for (int kb = 0; kb < 256; kb += 4) {
    v2f a = *reinterpret_cast<const v2f*>(zp + kb);
    v2f b = *reinterpret_cast<const v2f*>(w2p + kb);
    acc = WMMA_F32X4(a, b, acc);
  }
  __syncthreads();

  if (row < D) {                       // row acts as N (output feature) here
    const float bias2 = b2[row];
    #pragma unroll
    for (int r = 0; r < 8; ++r) ldsX[(r + half * 8) * 8 + row] = acc[r] + bias2;
  }
  __syncthreads();
  #pragma unroll
  for (int c = 0; c < D; ++c) out[c] = ldsX[row * 8 + c];
  __syncthreads();
}

// ---------------- one post-norm transformer (enc layer + dec layer + final norms) ----------------

template<int D>
__device__ void transformer16(const TP& p, const float* x, float* y,
                              float* ldsX, float* ldsZ, int lane) {
  float a[D], h[D], ffo[D], mem[D];
  // encoder layer
  attn_row<D>(x, p.t[0], p.t[1], p.t[2], p.t[3], a);
  #pragma unroll
  for (int c = 0; c < D; ++c) h[c] = x[c] + a[c];
  ln_row<D>(h, p.t[8], p.t[9]);                                  // e_n1
  ff16<D>(h, p.t[4], p.t[5], p.t[6], p.t[7], ffo, ldsX, ldsZ, lane);
  #pragma unroll
  for (int c = 0; c < D; ++c) mem[c] = h[c] + ffo[c];
  ln_row<D>(mem, p.t[10], p.t[11]);                              // e_n2
  ln_row<D>(mem, p.t[12], p.t[13]);                              // e_nf
  // decoder layer (tgt == src == x)
  attn_row<D>(x, p.t[14], p.t[15], p.t[16], p.t[17], a);         // self-attn
  #pragma unroll
  for (int c = 0; c < D; ++c) y[c] = x[c] + a[c];
  ln_row<D>(y, p.t[26], p.t[27]);                                // d_n1
  attn_row<D>(mem, p.t[18], p.t[19], p.t[20], p.t[21], a);       // cross-attn
  #pragma unroll
  for (int c = 0; c < D; ++c) y[c] += a[c];
  ln_row<D>(y, p.t[28], p.t[29]);                                // d_n2
  ff16<D>(y, p.t[22], p.t[23], p.t[24], p.t[25], ffo, ldsX, ldsZ, lane);
  #pragma unroll
  for (int c = 0; c < D; ++c) y[c] += ffo[c];
  ln_row<D>(y, p.t[30], p.t[31]);                                // d_n3
  ln_row<D>(y, p.t[32], p.t[33]);                                // d_nf
}

// ---------------- top-level kernel: one wave = one 16-row tile ----------------

__global__ __launch_bounds__(32)
void dognet_kernel(KArgs A) {
  __shared__ float ldsX[16 * 8];
  __shared__ float ldsZ[16 * 260];
  const int  lane  = threadIdx.x;
  const int  row   = lane & 15;
  const long gr    = (long)blockIdx.x * 16 + row;
  const bool valid = gr < (long)A.nrows;

  float x3[3], dogt[3], mant[3], x1[1], hpt[1], mpt[1];

  #pragma unroll
  for (int c = 0; c < 3; ++c) x3[c] = valid ? A.dog[gr * 3 + c] : 0.0f;
  transformer16<3>(A.p1, x3, dogt, ldsX, ldsZ, lane);

  #pragma unroll
  for (int c = 0; c < 3; ++c) x3[c] = valid ? A.man[gr * 3 + c] : 0.0f;
  transformer16<3>(A.p1, x3, mant, ldsX, ldsZ, lane);

  x1[0] = valid ? A.hp[gr] : 0.0f;
  transformer16<1>(A.p2, x1, hpt, ldsX, ldsZ, lane);
  x1[0] = valid ? A.mp[gr] : 0.0f;
  transformer16<1>(A.p2, x1, mpt, ldsX, ldsZ, lane);

  float q[8] = { dogt[0], dogt[1], dogt[2], mant[0], mant[1], mant[2], hpt[0], mpt[0] };
  float q3[8];
  transformer16<8>(A.p3, q, q3, ldsX, ldsZ, lane);

  if (valid && lane < 16) {
    #pragma unroll
    for (int c = 0; c < 3; ++c) {
      float m = A.l1b[c];
      #pragma unroll
      for (int k = 0; k < 8; ++k) m += q3[k] * A.l1w[c * 8 + k];
      A.out[gr * 3 + c] = m;                       // move  [B,3]
    }
    float bk = A.l2b[0];
    #pragma unroll
    for (int k = 0; k < 8; ++k) bk += q3[k] * A.l2w[k];
    A.out[(long)A.nrows * 3 + gr] = bk;            // bark  [B,1]
    A.out[(long)A.nrows * 4 + gr] = bk;            // shake [B,1]
  }
}

extern "C" void kernel_launch(void* const* d_in, const int* in_sizes, int n_in,
                              void* d_out, int out_size, void* d_ws, size_t ws_size,
                              hipStream_t stream) {
  (void)n_in; (void)out_size; (void)d_ws; (void)ws_size;
  KArgs A;
  A.dog = (const float*)d_in[0];
  A.man = (const float*)d_in[1];
  A.hp  = (const float*)d_in[2];
  A.mp  = (const float*)d_in[3];
  for (int i = 0; i < 34; ++i) A.p1.t[i] = (const float*)d_in[4 + i];    // p1 (d=3)
  for (int i = 0; i < 34; ++i) A.p2.t[i] = (const float*)d_in[38 + i];   // p2 (d=1)
  for (int i = 0; i < 34; ++i) A.p3.t[i] = (const float*)d_in[72 + i];   // p3 (d=8)
  A.l1w = (const float*)d_in[106];
  A.l1b = (const float*)d_in[107];
  A.l2w = (const float*)d_in[108];
  A.l2b = (const float*)d_in[109];
  A.out = (float*)d_out;
  A.nrows = in_sizes[2];                                                 // hp has B elements

  const int blocks = (A.nrows + 15) / 16;
  dognet_kernel<<<blocks, 32, 0, stream>>>(A);
}